// QuantumCircuit_65240553226500
// MI455X (gfx1250) — compile-verified
//
#include <hip/hip_runtime.h>
#include <math.h>

// MI455X / gfx1250: wave32, WMMA 16x16x4 f32 path.
// One workgroup (256 thr = 8 waves) simulates one sample; the 2^14 complex
// state lives entirely in LDS (128 KB of the WGP's 320 KB).

typedef float v2f __attribute__((ext_vector_type(2)));
typedef float v8f __attribute__((ext_vector_type(8)));

#define NQ       14
#define NSTATE   (1 << NQ)      // 16384
#define NLAYERS  4
#define NTHREADS 256
#define PI_F     3.14159265358979323846f

struct cplx { float r, i; };
__device__ __forceinline__ cplx cmul(cplx a, cplx b) {
    return { a.r * b.r - a.i * b.i, a.r * b.i + a.i * b.r };
}
__device__ __forceinline__ cplx cadd(cplx a, cplx b) { return { a.r + b.r, a.i + b.i }; }

// Fused single-qubit U = RZ(g) * RY(b) * RX(a)   (circuit applies RX, RY, RZ in order)
__device__ __forceinline__ void fusedU(float a, float b, float g, cplx U[4]) {
    float sa, ca, sb, cb, sg, cg;
    sincosf(0.5f * a, &sa, &ca);
    sincosf(0.5f * b, &sb, &cb);
    sincosf(0.5f * g, &sg, &cg);
    // M = RY * RX
    cplx m00 = {  cb * ca,  sb * sa };
    cplx m01 = { -sb * ca, -cb * sa };
    cplx m10 = {  sb * ca, -cb * sa };
    cplx m11 = {  cb * ca, -sb * sa };
    cplx p = { cg, -sg }, q = { cg, sg };   // RZ diag
    U[0] = cmul(p, m00); U[1] = cmul(p, m01);
    U[2] = cmul(q, m10); U[3] = cmul(q, m11);
}

__device__ __forceinline__ void encU(float theta, cplx U[4]) {  // real RY(theta)
    float s, c;
    sincosf(0.5f * theta, &s, &c);
    U[0] = { c, 0.f }; U[1] = { -s, 0.f };
    U[2] = { s, 0.f }; U[3] = {  c, 0.f };
}

// Build 16x16 complex G = U3 (x) U2 (x) U1 (x) U0 into LDS (row-major, G[m][k]).
// Local bit j of m/k corresponds to wire (q0 + j).
__device__ __forceinline__ void buildG(const cplx U[4][4], float* gr, float* gi) {
    const int t = threadIdx.x;           // 256 threads, one entry each
    const int m = t >> 4, k = t & 15;
    cplx v = U[0][(( m       & 1) << 1) | ( k       & 1)];
    v = cmul(v, U[1][(((m >> 1) & 1) << 1) | ((k >> 1) & 1)]);
    v = cmul(v, U[2][(((m >> 2) & 1) << 1) | ((k >> 2) & 1)]);
    v = cmul(v, U[3][(((m >> 3) & 1) << 1) | ((k >> 3) & 1)]);
    gr[m * 16 + k] = v.r;
    gi[m * 16 + k] = v.i;
}

// Apply 16x16 complex G to state bits [q0, q0+3]:  S' = G x S(16 x 1024).
// K=16 accumulated as 4 chunks of V_WMMA_F32_16X16X4_F32; complex via 4 real
// products: Re = Gr*Sr + (-Gi)*Si ; Im = Gr*Si + Gi*Sr.
__device__ __forceinline__ void applyBlock16(int q0, const float* gr, const float* gi,
                                             float* sre, float* sim) {
    const int tid  = threadIdx.x;
    const int lane = tid & 31;
    const int wv   = tid >> 5;          // wave 0..7
    const int h    = lane >> 4;         // half-wave selector
    const int mrow = lane & 15;

    // A fragments (16x4 chunks of G), layout: lane L: M=L&15, K={0,1}+2*(L>=16)
    v2f ar[4], aiv[4], an[4];
#pragma unroll
    for (int kk = 0; kk < 4; ++kk) {
        const int c0   = 4 * kk + 2 * h;
        const int base = mrow * 16 + c0;
        ar[kk].x  = gr[base]; ar[kk].y  = gr[base + 1];
        aiv[kk].x = gi[base]; aiv[kk].y = gi[base + 1];
        an[kk] = -aiv[kk];
    }

    const unsigned lomask = (1u << q0) - 1u;
    // 64 column tiles of 16; wave w owns tiles [8w, 8w+8)
#pragma unroll
    for (int tt = 0; tt < 8; ++tt) {
        const int      T     = wv * 8 + tt;
        const unsigned col   = (unsigned)(T * 16 + mrow);
        const unsigned base2 = ((col >> q0) << (q0 + 4)) | (col & lomask);

        v8f cre = { 0.f, 0.f, 0.f, 0.f, 0.f, 0.f, 0.f, 0.f };
        v8f cim = { 0.f, 0.f, 0.f, 0.f, 0.f, 0.f, 0.f, 0.f };
#pragma unroll
        for (int kk = 0; kk < 4; ++kk) {
            const unsigned krow = 4u * kk + 2u * h;         // B: lane L -> K rows {krow, krow+1}
            const unsigned iA = base2 + (krow << q0);
            const unsigned iB = base2 + ((krow + 1u) << q0);
            v2f br, bi;
            br.x = sre[iA]; br.y = sre[iB];
            bi.x = sim[iA]; bi.y = sim[iB];
            cre = __builtin_amdgcn_wmma_f32_16x16x4_f32(false, ar[kk],  false, br, (short)0, cre, false, false);
            cre = __builtin_amdgcn_wmma_f32_16x16x4_f32(false, an[kk],  false, bi, (short)0, cre, false, false);
            cim = __builtin_amdgcn_wmma_f32_16x16x4_f32(false, ar[kk],  false, bi, (short)0, cim, false, false);
            cim = __builtin_amdgcn_wmma_f32_16x16x4_f32(false, aiv[kk], false, br, (short)0, cim, false, false);
        }
        // D layout: VGPR j, this lane -> element (M = j + 8h, N = mrow); col = T*16 + N
#pragma unroll
        for (int j = 0; j < 8; ++j) {
            const unsigned M   = (unsigned)j + 8u * h;
            const unsigned idx = base2 + (M << q0);
            sre[idx] = cre[j];
            sim[idx] = cim[j];
        }
    }
}

// Generic 1-qubit gate on wire q (VALU pair update) — used for wires 12, 13.
__device__ __forceinline__ void apply1q(int q, const cplx U[4], float* sre, float* sim) {
    const unsigned mask = (1u << q) - 1u;
    for (unsigned t = threadIdx.x; t < (NSTATE / 2); t += NTHREADS) {
        const unsigned i0 = ((t & ~mask) << 1) | (t & mask);
        const unsigned i1 = i0 | (1u << q);
        cplx a0 = { sre[i0], sim[i0] }, a1 = { sre[i1], sim[i1] };
        cplx n0 = cadd(cmul(U[0], a0), cmul(U[1], a1));
        cplx n1 = cadd(cmul(U[2], a0), cmul(U[3], a1));
        sre[i0] = n0.r; sim[i0] = n0.i;
        sre[i1] = n1.r; sim[i1] = n1.i;
    }
}

// CNOT(c,t) followed by RZ(theta) on t — a phase-permutation, done as one pass.
__device__ __forceinline__ void cnotrz(int c, int t, float theta, float* sre, float* sim) {
    float sv, cv;
    sincosf(0.5f * theta, &sv, &cv);
    const cplx P = { cv, -sv }, Q = { cv, sv };
    const unsigned mask = (1u << t) - 1u;
    for (unsigned p = threadIdx.x; p < (NSTATE / 2); p += NTHREADS) {
        const unsigned i0 = ((p & ~mask) << 1) | (p & mask);
        const unsigned i1 = i0 | (1u << t);
        const bool ctrl = (i0 >> c) & 1u;
        cplx a0 = { sre[i0], sim[i0] }, a1 = { sre[i1], sim[i1] };
        cplx s0 = ctrl ? a1 : a0;       // CNOT: swap target pair where control=1
        cplx s1 = ctrl ? a0 : a1;
        cplx n0 = cmul(P, s0), n1 = cmul(Q, s1);
        sre[i0] = n0.r; sim[i0] = n0.i;
        sre[i1] = n1.r; sim[i1] = n1.i;
    }
}

__global__ __launch_bounds__(NTHREADS)
void qsim_kernel(const float* __restrict__ x,
                 const float* __restrict__ prx, const float* __restrict__ pry,
                 const float* __restrict__ prz, const float* __restrict__ pent,
                 float* __restrict__ out) {
    extern __shared__ float smem[];
    float* sre  = smem;                 // 16384
    float* sim  = sre + NSTATE;         // 16384
    float* gr   = sim + NSTATE;         // 256
    float* gi   = gr + 256;             // 256
    float* obuf = gi + 256;             // 16

    const int s   = blockIdx.x;
    const int tid = threadIdx.x;

    // |0...0> init
    for (int k = tid; k < NSTATE; k += NTHREADS) { sre[k] = 0.f; sim[k] = 0.f; }
    if (tid < 16) obuf[tid] = 0.f;
    if (tid == 0) sre[0] = 1.f;
    __syncthreads();

    cplx Uq[4][4];

    // --- Angle encoding: RY(tanh(x)*pi) per wire ---
    for (int b = 0; b < 3; ++b) {                   // wires 0..11 via WMMA blocks
#pragma unroll
        for (int j = 0; j < 4; ++j)
            encU(tanhf(x[s * NQ + 4 * b + j]) * PI_F, Uq[j]);
        buildG(Uq, gr, gi);
        __syncthreads();
        applyBlock16(4 * b, gr, gi, sre, sim);
        __syncthreads();
    }
    for (int q = 12; q < 14; ++q) {                 // wires 12,13 via VALU
        cplx Ue[4];
        encU(tanhf(x[s * NQ + q]) * PI_F, Ue);
        apply1q(q, Ue, sre, sim);
        __syncthreads();
    }

    // --- Variational layers ---
    for (int l = 0; l < NLAYERS; ++l) {
        for (int b = 0; b < 3; ++b) {               // fused RZ*RY*RX blocks via WMMA
#pragma unroll
            for (int j = 0; j < 4; ++j) {
                const int q = 4 * b + j;
                fusedU(prx[l * NQ + q], pry[l * NQ + q], prz[l * NQ + q], Uq[j]);
            }
            buildG(Uq, gr, gi);
            __syncthreads();
            applyBlock16(4 * b, gr, gi, sre, sim);
            __syncthreads();
        }
        for (int q = 12; q < 14; ++q) {
            cplx Uf[4];
            fusedU(prx[l * NQ + q], pry[l * NQ + q], prz[l * NQ + q], Uf);
            apply1q(q, Uf, sre, sim);
            __syncthreads();
        }
        for (int i = 0; i < NQ - 1; ++i) {          // entangler chain, program order
            cnotrz(i, i + 1, pent[l * (NQ - 1) + i], sre, sim);
            __syncthreads();
        }
    }

    // --- <Z_i> = sum_k (1 - 2*bit_i(k)) * |amp_k|^2 ---
    float acc[NQ];
#pragma unroll
    for (int i = 0; i < NQ; ++i) acc[i] = 0.f;
    for (int k = tid; k < NSTATE; k += NTHREADS) {
        const float re = sre[k], im = sim[k];
        const float p = re * re + im * im;
#pragma unroll
        for (int i = 0; i < NQ; ++i) acc[i] += ((k >> i) & 1) ? -p : p;
    }
#pragma unroll
    for (int i = 0; i < NQ; ++i) {
        float v = acc[i];
        for (int off = 16; off > 0; off >>= 1) v += __shfl_down(v, off, 32);  // wave32
        if ((tid & 31) == 0) atomicAdd(&obuf[i], v);
    }
    __syncthreads();
    if (tid < NQ) out[s * NQ + tid] = obuf[tid];
}

extern "C" void kernel_launch(void* const* d_in, const int* in_sizes, int n_in,
                              void* d_out, int out_size, void* d_ws, size_t ws_size,
                              hipStream_t stream) {
    (void)in_sizes; (void)n_in; (void)out_size; (void)d_ws; (void)ws_size;
    const float* x    = (const float*)d_in[0];
    const float* prx  = (const float*)d_in[1];
    const float* pry  = (const float*)d_in[2];
    const float* prz  = (const float*)d_in[3];
    const float* pent = (const float*)d_in[4];
    float* out = (float*)d_out;

    const size_t shmem = (size_t)(2 * NSTATE + 2 * 256 + 16) * sizeof(float); // 133184 B
    (void)hipFuncSetAttribute((const void*)qsim_kernel,
                              hipFuncAttributeMaxDynamicSharedMemorySize, (int)shmem);
    qsim_kernel<<<dim3(128), dim3(NTHREADS), shmem, stream>>>(x, prx, pry, prz, pent, out);
}